// MoEGate_20426864460257
// MI455X (gfx1250) — compile-verified
//
#include <hip/hip_runtime.h>
#include <math.h>

typedef __attribute__((ext_vector_type(2))) float v2f;
typedef __attribute__((ext_vector_type(4))) float v4f;
typedef __attribute__((ext_vector_type(8))) float v8f;

#define TOPK      8
#define NEXP      64
#define DIM       2048
#define SEQ       8192
#define BSZ       4
#define TOKENS    (BSZ * SEQ)
#define KC        64
#define KCP       (KC + 4)          // padded LDS row stride (dwords): 68 -> lane stride 4 banks, conflict-free
#define NCHUNK    (DIM / KC)
#define TOK_WG    128               // tokens per workgroup (8 waves x 16 rows)
#define SCP       65                // score row stride (dwords): stride-1 bank walk for row scans
#define ALPHA     0.1f

// ---------------------------------------------------------------------------
// Async global -> LDS b128 copy (ASYNCcnt-tracked), GVS addressing mode:
//   mem = SADDR(64b SGPR) + VADDR(32b byte offset) + IOFFSET
//   VDST = wave-relative LDS byte address
// ---------------------------------------------------------------------------
__device__ __forceinline__ void async_copy_b128(const float* lds_dst,
                                                const float* gbase,
                                                unsigned gbyte_off) {
    unsigned lds_off = (unsigned)(uintptr_t)lds_dst;   // low 32 bits of generic LDS ptr = LDS offset
    asm volatile("global_load_async_to_lds_b128 %0, %1, %2 offset:0"
                 :: "v"(lds_off), "v"(gbyte_off), "s"(gbase)
                 : "memory");
}

// Issue one K-chunk of staging: x tile [TOK_WG x KC] + W chunk [NEXP x KC].
// 12 async b128 ops per thread (8 for x, 4 for W).
#define ASYNC_OPS_PER_THREAD 12
__device__ __forceinline__ void issue_chunk_async(const float* __restrict__ xbase,
                                                  const float* __restrict__ wbase,
                                                  float* __restrict__ ldsx,
                                                  float* __restrict__ ldsw,
                                                  int tid) {
    #pragma unroll
    for (int i = 0; i < 8; ++i) {                 // 128*64/4 = 2048 xfers / 256 thr
        int idx = tid + i * 256;
        int r = idx >> 4;                          // / (KC/4)
        int c = (idx & 15) << 2;
        async_copy_b128(ldsx + r * KCP + c, xbase, (unsigned)((r * DIM + c) * sizeof(float)));
    }
    #pragma unroll
    for (int i = 0; i < 4; ++i) {                 // 64*64/4 = 1024 xfers / 256 thr
        int idx = tid + i * 256;
        int r = idx >> 4;
        int c = (idx & 15) << 2;
        async_copy_b128(ldsw + r * KCP + c, wbase, (unsigned)((r * DIM + c) * sizeof(float)));
    }
}

// ---------------------------------------------------------------------------
// Workspace init: zero per-(batch,expert) histogram + score-sum accumulators.
// ---------------------------------------------------------------------------
__global__ void moe_ws_init(float* ws) {
    int i = threadIdx.x;
    if (i < 2 * BSZ * NEXP) ws[i] = 0.0f;
}

// ---------------------------------------------------------------------------
// Fused gate kernel: double-buffered async staging + WMMA fp32 GEMM
// (x @ W^T) + softmax + top-8 + aux statistics.
// One block = 128 tokens (one batch). 8 waves, each wave computes a
// 16(token) x 64(expert) strip with V_WMMA_F32_16X16X4_F32.
// ---------------------------------------------------------------------------
__global__ __launch_bounds__(256) void moe_gate_kernel(
    const float* __restrict__ x,      // [TOKENS, DIM]
    const float* __restrict__ w,      // [NEXP, DIM]
    float* __restrict__ out_idx,      // [TOKENS, TOPK]  (indices as float)
    float* __restrict__ out_wgt,      // [TOKENS, TOPK]
    float* __restrict__ g_cnt,        // [BSZ, NEXP] top-k hit counts
    float* __restrict__ g_ssum)       // [BSZ, NEXP] softmax score sums
{
    __shared__ float lds_x[2][TOK_WG * KCP];   // 2 x 34.8 KB
    __shared__ float lds_w[2][NEXP * KCP];     // 2 x 17.4 KB
    __shared__ float lds_sc[TOK_WG * SCP];     // 33.3 KB (logits, then scores)
    __shared__ float shist[NEXP];

    const int tid  = threadIdx.x;
    const int wave = tid >> 5;
    const int lane = tid & 31;
    const size_t tok0 = (size_t)blockIdx.x * TOK_WG;
    const int wtok = wave * 16;

    // WMMA fp32 16x16x4 fragment addressing (wave32):
    //   A: lanes 0-15 hold {K=0,K=1} of row M=lane; lanes 16-31 hold {K=2,K=3}
    //   B: lanes 0-15 hold {K=0,K=1} of col N=lane; lanes 16-31 hold {K=2,K=3}
    const int mrow = lane & 15;
    const int koff = (lane >> 4) * 2;

    v8f acc0 = {}, acc1 = {}, acc2 = {}, acc3 = {};

    const float* xb = x + tok0 * DIM;

    // Prologue: start chunk 0 into buffer 0.
    issue_chunk_async(xb, w, lds_x[0], lds_w[0], tid);

    for (int ic = 0; ic < NCHUNK; ++ic) {
        const int cur = ic & 1;
        const int nxt = cur ^ 1;

        __syncthreads();   // everyone finished reading buf[nxt] (iteration ic-1)

        if (ic + 1 < NCHUNK) {
            issue_chunk_async(xb + (ic + 1) * KC, w + (ic + 1) * KC,
                              lds_x[nxt], lds_w[nxt], tid);
            // 12 newest ops (chunk ic+1) may stay outstanding; wait for chunk ic.
            asm volatile("s_wait_asynccnt 12" ::: "memory");
        } else {
            asm volatile("s_wait_asynccnt 0" ::: "memory");
        }
        __syncthreads();   // buf[cur] complete in all waves

        if (ic + 2 < NCHUNK) {  // warm L2 two chunks ahead (global_prefetch_b8)
            const float* nx = xb + (size_t)(tid >> 1) * DIM + (ic + 2) * KC + (tid & 1) * 32;
            __builtin_prefetch(nx, 0, 2);
        }

        const float* bx = lds_x[cur];
        const float* bw = lds_w[cur];

        #pragma unroll 4
        for (int k = 0; k < KC; k += 4) {
            v2f a  = *(const v2f*)(bx + (wtok + mrow) * KCP + k + koff);
            v2f b0 = *(const v2f*)(bw + ( 0 + mrow) * KCP + k + koff);
            v2f b1 = *(const v2f*)(bw + (16 + mrow) * KCP + k + koff);
            v2f b2 = *(const v2f*)(bw + (32 + mrow) * KCP + k + koff);
            v2f b3 = *(const v2f*)(bw + (48 + mrow) * KCP + k + koff);
            acc0 = __builtin_amdgcn_wmma_f32_16x16x4_f32(false, a, false, b0, (short)0, acc0, false, false);
            acc1 = __builtin_amdgcn_wmma_f32_16x16x4_f32(false, a, false, b1, (short)0, acc1, false, false);
            acc2 = __builtin_amdgcn_wmma_f32_16x16x4_f32(false, a, false, b2, (short)0, acc2, false, false);
            acc3 = __builtin_amdgcn_wmma_f32_16x16x4_f32(false, a, false, b3, (short)0, acc3, false, false);
        }
    }

    // --- spill logits to LDS ---
    // C/D layout: lanes 0-15: (M=vg, N=lane); lanes 16-31: (M=vg+8, N=lane-16)
    {
        const int eh = lane & 15;
        const int mbase = (lane >> 4) * 8;
        #pragma unroll
        for (int vg = 0; vg < 8; ++vg) {
            int row = wtok + mbase + vg;
            lds_sc[row * SCP +  0 + eh] = acc0[vg];
            lds_sc[row * SCP + 16 + eh] = acc1[vg];
            lds_sc[row * SCP + 32 + eh] = acc2[vg];
            lds_sc[row * SCP + 48 + eh] = acc3[vg];
        }
    }
    __syncthreads();
    if (tid < NEXP) shist[tid] = 0.0f;
    __syncthreads();

    // --- per-token softmax + top-8 (one thread per token, scores kept in LDS) ---
    if (tid < TOK_WG) {
        float* row = &lds_sc[tid * SCP];
        float mx = -3.0e38f;
        for (int e = 0; e < NEXP; ++e) mx = fmaxf(mx, row[e]);
        float sum = 0.0f;
        for (int e = 0; e < NEXP; ++e) { float v = __expf(row[e] - mx); row[e] = v; sum += v; }
        float inv = 1.0f / sum;
        for (int e = 0; e < NEXP; ++e) row[e] *= inv;   // normalized scores (kept for aux)

        unsigned long long used = 0ull;
        float tw[TOPK]; int ti[TOPK]; float wsum = 0.0f;
        for (int k = 0; k < TOPK; ++k) {
            float bv = -1.0f; int bi = 0;
            for (int e = 0; e < NEXP; ++e) {
                float v = row[e];
                bool ok = (((used >> e) & 1ull) == 0ull) && (v > bv);
                bv = ok ? v : bv;
                bi = ok ? e : bi;
            }
            used |= (1ull << bi);
            tw[k] = bv; ti[k] = bi; wsum += bv;
        }
        float wn = 1.0f / (wsum + 1e-20f);
        size_t gt = tok0 + tid;
        #pragma unroll
        for (int k = 0; k < TOPK; ++k) {
            out_idx[gt * TOPK + k] = (float)ti[k];
            out_wgt[gt * TOPK + k] = tw[k] * wn;
            atomicAdd(&shist[ti[k]], 1.0f);   // ds_add_f32
        }
    }
    __syncthreads();

    // --- per-block reduction -> one global atomic per (batch, expert) ---
    if (tid < NEXP) {
        const int b = (int)(tok0 / SEQ);
        float colsum = 0.0f;
        for (int r = 0; r < TOK_WG; ++r) colsum += lds_sc[r * SCP + tid];
        atomicAdd(&g_ssum[b * NEXP + tid], colsum);
        atomicAdd(&g_cnt [b * NEXP + tid], shist[tid]);
    }
}

// ---------------------------------------------------------------------------
// Final reduction: aux = ALPHA * mean_b( sum_e ce*mean_scores )
//   ce = cnt * NEXP/(SEQ*TOPK);  mean_scores = ssum/SEQ
// ---------------------------------------------------------------------------
__global__ void moe_aux_kernel(const float* __restrict__ g_cnt,
                               const float* __restrict__ g_ssum,
                               float* __restrict__ aux_out)
{
    __shared__ float red[BSZ * NEXP];
    int i = threadIdx.x;                           // 256 = BSZ*NEXP
    red[i] = g_cnt[i] * g_ssum[i];
    __syncthreads();
    for (int s = (BSZ * NEXP) / 2; s > 0; s >>= 1) {
        if (i < s) red[i] += red[i + s];
        __syncthreads();
    }
    if (i == 0) {
        const float scale = ALPHA * ((float)NEXP / ((float)SEQ * (float)TOPK))
                            / (float)SEQ / (float)BSZ;
        aux_out[0] = red[0] * scale;
    }
}

extern "C" void kernel_launch(void* const* d_in, const int* in_sizes, int n_in,
                              void* d_out, int out_size, void* d_ws, size_t ws_size,
                              hipStream_t stream) {
    const float* x = (const float*)d_in[0];   // [4, 8192, 2048]
    const float* w = (const float*)d_in[1];   // [64, 2048]
    float* out = (float*)d_out;
    float* out_idx = out;                              // [T, 8] indices as float
    float* out_wgt = out + (size_t)TOKENS * TOPK;      // [T, 8]
    float* aux     = out + (size_t)2 * TOKENS * TOPK;  // scalar

    float* g_cnt  = (float*)d_ws;              // [BSZ, NEXP]
    float* g_ssum = g_cnt + BSZ * NEXP;        // [BSZ, NEXP]

    moe_ws_init<<<1, 512, 0, stream>>>(g_cnt);
    moe_gate_kernel<<<TOKENS / TOK_WG, 256, 0, stream>>>(x, w, out_idx, out_wgt, g_cnt, g_ssum);
    moe_aux_kernel<<<1, BSZ * NEXP, 0, stream>>>(g_cnt, g_ssum, aux);
}